// GCN_25469156065531
// MI455X (gfx1250) — compile-verified
//
#include <hip/hip_runtime.h>
#include <math.h>

#define N_NODES 100000
#define N_EDGES 6400000

typedef float v2f __attribute__((ext_vector_type(2)));
typedef float v4f __attribute__((ext_vector_type(4)));
typedef float v8f __attribute__((ext_vector_type(8)));

// ---------------- degree / normalization ----------------

__global__ void gcn_init_deg(float* __restrict__ deg, int n) {
    int i = blockIdx.x * blockDim.x + threadIdx.x;
    if (i < n) deg[i] = 1.0f;  // self-loop contributes 1 to every node
}

__global__ void gcn_degree(const int* __restrict__ col, float* __restrict__ deg, int e) {
    int i = blockIdx.x * blockDim.x + threadIdx.x;
    if (i < e) atomicAdd(&deg[col[i]], 1.0f);
}

__global__ void gcn_rsqrt_inplace(float* __restrict__ d, int n) {
    int i = blockIdx.x * blockDim.x + threadIdx.x;
    if (i < n) {
        float v = d[i];
        d[i] = (v > 0.0f) ? rsqrtf(v) : 0.0f;
    }
}

// ---------------- layer 1 GEMM: X[100000,128] @ W1[128,4] via WMMA f32 16x16x4 ----
// One wave per 16-row tile; B = W1 zero-padded to 16 columns, staged in LDS
// transposed [n][k] (stride 132 to avoid bank conflicts).
// K-order remap (A and B agree, reduction commutative): lane half hi covers
// physical k in [64*hi, 64*hi+63]; WMMA step t uses k = 64*hi + 2t + j for
// VGPR slot j. So one b128 load (4 contiguous floats) feeds two WMMA steps.
// A layout (16x4 f32): lane l holds row (l&15); vgpr j holds K-slot j+2*(l>>4).
// C/D layout (16x16 f32): vgpr j, lane l -> row = j + 8*(l>>4), col = l&15.

#define WSTRIDE 132

__global__ void gcn_gemm_xw1_wmma(const float* __restrict__ x,
                                  const float* __restrict__ W,   // [128,4] row-major
                                  float* __restrict__ h,         // [n,4]
                                  int n_nodes) {
    __shared__ float Wpad[16 * WSTRIDE];   // [n=16][k=128] transposed, padded stride

    // cooperative staging of zero-padded, transposed W1
    for (int idx = threadIdx.x; idx < 16 * 128; idx += blockDim.x) {
        int nn = idx >> 7;     // output column 0..15
        int kk = idx & 127;    // k 0..127
        Wpad[nn * WSTRIDE + kk] = (nn < 4) ? W[kk * 4 + nn] : 0.0f;
    }
    __syncthreads();

    const int lane = threadIdx.x & 31;
    const int wave = threadIdx.x >> 5;
    const int tile = blockIdx.x * (blockDim.x >> 5) + wave;
    if (tile * 16 >= n_nodes) return;      // wave-uniform: EXEC all-ones at WMMA

    const int m  = lane & 15;              // A row within tile; B/C column
    const int hi = lane >> 4;              // K-half selector
    const float* __restrict__ xrow = x + (size_t)(tile * 16 + m) * 128 + hi * 64;
    const float* __restrict__ wrow = Wpad + m * WSTRIDE + hi * 64;

    v8f c = {};
    #pragma unroll 4
    for (int s = 0; s < 32; s += 2) {      // two WMMA steps per iteration
        v4f av = *(const v4f*)(xrow + 2 * s);   // global_load_b128
        v4f bv = *(const v4f*)(wrow + 2 * s);   // ds_load_b128
        v2f a0 = { av.x, av.y };
        v2f a1 = { av.z, av.w };
        v2f b0 = { bv.x, bv.y };
        v2f b1 = { bv.z, bv.w };
        // (neg_a, A, neg_b, B, c_mod, C, reuse_a, reuse_b)
        c = __builtin_amdgcn_wmma_f32_16x16x4_f32(false, a0, false, b0, (short)0, c,
                                                  false, false);
        c = __builtin_amdgcn_wmma_f32_16x16x4_f32(false, a1, false, b1, (short)0, c,
                                                  false, false);
    }

    if (m < 4) {
        #pragma unroll
        for (int j = 0; j < 8; ++j) {
            int r = j + 8 * hi;
            h[(size_t)(tile * 16 + r) * 4 + m] = c[j];
        }
    }
}

// ---------------- tiny per-node matmuls (layers 2/3, classifier) ----------------

template <int FI, int FO>
__global__ void gcn_node_matmul(const float* __restrict__ x, const float* __restrict__ W,
                                float* __restrict__ h, int n) {
    int i = blockIdx.x * blockDim.x + threadIdx.x;
    if (i >= n) return;
    float in[FI];
    #pragma unroll
    for (int f = 0; f < FI; ++f) in[f] = x[(size_t)i * FI + f];
    #pragma unroll
    for (int o = 0; o < FO; ++o) {
        float s = 0.0f;
        #pragma unroll
        for (int f = 0; f < FI; ++f) s += in[f] * W[f * FO + o];
        h[(size_t)i * FO + o] = s;
    }
}

__global__ void gcn_classifier(const float* __restrict__ y3, const float* __restrict__ Wc,
                               const float* __restrict__ bc, float* __restrict__ out, int n) {
    int i = blockIdx.x * blockDim.x + threadIdx.x;
    if (i >= n) return;
    float a0 = y3[(size_t)i * 2 + 0];
    float a1 = y3[(size_t)i * 2 + 1];
    #pragma unroll
    for (int j = 0; j < 4; ++j)
        out[(size_t)i * 4 + j] = a0 * Wc[j] + a1 * Wc[4 + j] + bc[j];
}

// ---------------- propagation ----------------

__global__ void gcn_zero(float* __restrict__ p, int count) {
    int i = blockIdx.x * blockDim.x + threadIdx.x;
    if (i < count) p[i] = 0.0f;
}

template <int F>
__global__ void gcn_edge_scatter(const int* __restrict__ row, const int* __restrict__ col,
                                 const float* __restrict__ dinv, const float* __restrict__ h,
                                 float* __restrict__ agg, int e) {
    int i = blockIdx.x * blockDim.x + threadIdx.x;
    if (i >= e) return;
    int r = row[i];
    int c = col[i];
    float w = dinv[r] * dinv[c];
    #pragma unroll
    for (int f = 0; f < F; ++f)
        atomicAdd(&agg[(size_t)c * F + f], w * h[(size_t)r * F + f]);
}

// fused: self-loop contribution + bias + ReLU
template <int F>
__global__ void gcn_finalize(const float* __restrict__ agg, const float* __restrict__ h,
                             const float* __restrict__ dinv, const float* __restrict__ b,
                             float* __restrict__ y, int n) {
    int i = blockIdx.x * blockDim.x + threadIdx.x;
    if (i >= n) return;
    float w = dinv[i] * dinv[i];
    #pragma unroll
    for (int f = 0; f < F; ++f) {
        float v = agg[(size_t)i * F + f] + w * h[(size_t)i * F + f] + b[f];
        y[(size_t)i * F + f] = (v > 0.0f) ? v : 0.0f;
    }
}

// ---------------- host orchestration ----------------

extern "C" void kernel_launch(void* const* d_in, const int* in_sizes, int n_in,
                              void* d_out, int out_size, void* d_ws, size_t ws_size,
                              hipStream_t stream) {
    (void)in_sizes; (void)n_in; (void)out_size; (void)ws_size;

    const float* x  = (const float*)d_in[0];
    const int*   ei = (const int*)d_in[1];     // [2, E] int32
    const float* W1 = (const float*)d_in[2];
    const float* b1 = (const float*)d_in[3];
    const float* W2 = (const float*)d_in[4];
    const float* b2 = (const float*)d_in[5];
    const float* W3 = (const float*)d_in[6];
    const float* b3 = (const float*)d_in[7];
    const float* Wc = (const float*)d_in[8];
    const float* bc = (const float*)d_in[9];

    const int n = N_NODES;
    const int e = N_EDGES;
    const int* row = ei;
    const int* col = ei + e;

    // workspace layout (floats)
    float* ws   = (float*)d_ws;
    float* dinv = ws;                 // [n]      (holds deg, then deg^-1/2 in place)
    float* h    = dinv + n;           // [n,4]    pre-aggregation features
    float* agg  = h + 4 * n;          // [n,4]
    float* y1   = agg + 4 * n;        // [n,4]
    float* y2   = y1 + 4 * n;         // [n,4]

    float* out = (float*)d_out;       // [n,4]
    float* y3  = out + 4 * n;         // [n,2]  (second tuple element)

    const int T = 256;
    const int gN  = (n + T - 1) / T;
    const int gN4 = (4 * n + T - 1) / T;
    const int gE  = (e + T - 1) / T;
    const int tiles = n / 16;                          // 6250, exact
    const int gGemm = (tiles + (T / 32) - 1) / (T / 32);

    // normalization: deg = 1 + in-degree ; dinv = rsqrt(deg)
    gcn_init_deg<<<gN, T, 0, stream>>>(dinv, n);
    gcn_degree<<<gE, T, 0, stream>>>(col, dinv, e);
    gcn_rsqrt_inplace<<<gN, T, 0, stream>>>(dinv, n);

    // ---- layer 1: h = X @ W1 (WMMA) ; propagate ; y1 = relu ----
    gcn_gemm_xw1_wmma<<<gGemm, T, 0, stream>>>(x, W1, h, n);
    gcn_zero<<<gN4, T, 0, stream>>>(agg, 4 * n);
    gcn_edge_scatter<4><<<gE, T, 0, stream>>>(row, col, dinv, h, agg, e);
    gcn_finalize<4><<<gN, T, 0, stream>>>(agg, h, dinv, b1, y1, n);

    // ---- layer 2: h = y1 @ W2 ; propagate ; y2 = relu ----
    gcn_node_matmul<4, 4><<<gN, T, 0, stream>>>(y1, W2, h, n);
    gcn_zero<<<gN4, T, 0, stream>>>(agg, 4 * n);
    gcn_edge_scatter<4><<<gE, T, 0, stream>>>(row, col, dinv, h, agg, e);
    gcn_finalize<4><<<gN, T, 0, stream>>>(agg, h, dinv, b2, y2, n);

    // ---- layer 3: h = y2 @ W3 (4->2) ; propagate ; y3 = relu ----
    gcn_node_matmul<4, 2><<<gN, T, 0, stream>>>(y2, W3, h, n);
    gcn_zero<<<gN4, T, 0, stream>>>(agg, 2 * n);       // only first 2n used
    gcn_edge_scatter<2><<<gE, T, 0, stream>>>(row, col, dinv, h, agg, e);
    gcn_finalize<2><<<gN, T, 0, stream>>>(agg, h, dinv, b3, y3, n);

    // ---- classifier: out = y3 @ Wc + bc ----
    gcn_classifier<<<gN, T, 0, stream>>>(y3, Wc, bc, out, n);
}